// TestModel_42777874268657
// MI455X (gfx1250) — compile-verified
//
#include <hip/hip_runtime.h>
#include <stdint.h>

typedef __attribute__((ext_vector_type(2))) float v2f;
typedef __attribute__((ext_vector_type(8))) float v8f;
typedef unsigned long long u64;
typedef unsigned int u32;

#define N_IN   4096
#define M_ALL  8192
#define NWORDS 128            // 8192 / 64 bits
#define SORT_THREADS 1024

// ---------------------------------------------------------------------------
// 1) Prep: hflip view 2, concat, threshold, build sort key source
// ---------------------------------------------------------------------------
__global__ void prep_kernel(const float* __restrict__ boxes1,
                            const float* __restrict__ boxes2,
                            const float* __restrict__ scores1,
                            const float* __restrict__ scores2,
                            const int*   __restrict__ labels1,
                            const int*   __restrict__ labels2,
                            const int*   __restrict__ widthp,
                            float* __restrict__ boxesAll,
                            float* __restrict__ scoresAll,
                            int*   __restrict__ labelsAll,
                            int*   __restrict__ validAll,
                            float* __restrict__ sArr) {
  int i = blockIdx.x * blockDim.x + threadIdx.x;
  if (i >= M_ALL) return;
  float w = (float)(*widthp);
  float x1, y1, x2, y2, sc; int lb;
  if (i < N_IN) {
    x1 = boxes1[i * 4 + 0]; y1 = boxes1[i * 4 + 1];
    x2 = boxes1[i * 4 + 2]; y2 = boxes1[i * 4 + 3];
    sc = scores1[i]; lb = labels1[i];
  } else {
    int j = i - N_IN;
    x1 = w - boxes2[j * 4 + 2]; y1 = boxes2[j * 4 + 1];
    x2 = w - boxes2[j * 4 + 0]; y2 = boxes2[j * 4 + 3];
    sc = scores2[j]; lb = labels2[j];
  }
  boxesAll[i * 4 + 0] = x1; boxesAll[i * 4 + 1] = y1;
  boxesAll[i * 4 + 2] = x2; boxesAll[i * 4 + 3] = y2;
  scoresAll[i] = sc;
  labelsAll[i] = lb;
  int v = (sc >= 0.5f) ? 1 : 0;
  validAll[i] = v;
  sArr[i] = v ? sc : -1.0f;
}

// ---------------------------------------------------------------------------
// 2) Bitonic sort of 8192 u64 keys in LDS (single workgroup, 64KB dyn LDS).
//    key = (~ord(s) << 32) | idx  -> ascending sort == descending score,
//    stable on ties (idx tie-break), matching jnp.argsort(-s).
// ---------------------------------------------------------------------------
__global__ void __launch_bounds__(SORT_THREADS) sort_kernel(
    const float* __restrict__ sArr, u32* __restrict__ order) {
  extern __shared__ u64 smem[];
  int tid = threadIdx.x;
  for (int i = tid; i < M_ALL; i += SORT_THREADS) {
    u32 b = __float_as_uint(sArr[i]);
    u32 ord = b ^ ((b >> 31) ? 0xFFFFFFFFu : 0x80000000u);  // monotone f32->u32
    u32 key = ~ord;                                         // ascending == desc score
    smem[i] = ((u64)key << 32) | (u32)i;
  }
  __syncthreads();
  for (int k = 2; k <= M_ALL; k <<= 1) {
    for (int j = k >> 1; j > 0; j >>= 1) {
      for (int i = tid; i < M_ALL; i += SORT_THREADS) {
        int ixj = i ^ j;
        if (ixj > i) {
          bool up = ((i & k) == 0);
          u64 a = smem[i], c = smem[ixj];
          bool swap = up ? (a > c) : (a < c);
          if (swap) { smem[i] = c; smem[ixj] = a; }
        }
      }
      __syncthreads();
    }
  }
  for (int i = tid; i < M_ALL; i += SORT_THREADS)
    order[i] = (u32)(smem[i] & 0xFFFFFFFFu);
}

// ---------------------------------------------------------------------------
// 3) Gather sorted arrays
// ---------------------------------------------------------------------------
__global__ void gather_kernel(const u32* __restrict__ order,
                              const float* __restrict__ boxesAll,
                              const float* __restrict__ scoresAll,
                              const int* __restrict__ labelsAll,
                              const int* __restrict__ validAll,
                              float* __restrict__ sb, float* __restrict__ ss,
                              int* __restrict__ sl, int* __restrict__ sv) {
  int i = blockIdx.x * blockDim.x + threadIdx.x;
  if (i >= M_ALL) return;
  u32 o = order[i];
  sb[i * 4 + 0] = boxesAll[o * 4 + 0];
  sb[i * 4 + 1] = boxesAll[o * 4 + 1];
  sb[i * 4 + 2] = boxesAll[o * 4 + 2];
  sb[i * 4 + 3] = boxesAll[o * 4 + 3];
  ss[i] = scoresAll[o];
  sl[i] = labelsAll[o];
  sv[i] = validAll[o];
}

// ---------------------------------------------------------------------------
// 4) Suppression bitmask. One wave32 block per (16-row x 64-col) tile.
//    area_i + area_j outer-sum per 16x16 sub-tile via V_WMMA_F32_16X16X4_F32:
//      A[m][:] = [area_row[m], 1, 0, 0],  B[:][n] = [1, area_col[n], 0, 0]^T
//      => D[m][n] = area_row[m] + area_col[n]
//    Bit pack via wave32 ballot (lanes 0-15 = M=r, lanes 16-31 = M=r+8).
// ---------------------------------------------------------------------------
__global__ void __launch_bounds__(32) mask_kernel(
    const float* __restrict__ sb, const int* __restrict__ sl,
    u64* __restrict__ mask) {
  __shared__ float rbx[16][4];
  __shared__ int   rlb[16];
  int lane = threadIdx.x;
  int tr = blockIdx.y;            // row tile (16 rows)
  int cw = blockIdx.x;            // 64-bit column word
  int rowBase = tr * 16;

  if (lane < 16) {
    int r = rowBase + lane;
    rbx[lane][0] = sb[r * 4 + 0]; rbx[lane][1] = sb[r * 4 + 1];
    rbx[lane][2] = sb[r * 4 + 2]; rbx[lane][3] = sb[r * 4 + 3];
    rlb[lane] = sl[r];
  }
  __syncthreads();

  bool lo = lane < 16;
  int n = lane & 15;
  int li = n;  // safe LDS index for all lanes
  float rArea = (rbx[li][2] - rbx[li][0]) * (rbx[li][3] - rbx[li][1]);

  // A-matrix 16x4 f32 layout: VGPR0 = K0 (lanes 0-15) / K2 (16-31);
  //                           VGPR1 = K1 / K3. K2,K3 are zero.
  v2f A;
  A.x = lo ? rArea : 0.0f;
  A.y = lo ? 1.0f  : 0.0f;

  u64 word = 0ull;
#pragma unroll
  for (int st = 0; st < 4; ++st) {
    int col = cw * 64 + st * 16 + n;
    __builtin_prefetch(sb + (u64)(cw * 64 + ((st + 1) & 3) * 16 + n) * 4, 0, 1);
    float cx1 = sb[col * 4 + 0], cy1 = sb[col * 4 + 1];
    float cx2 = sb[col * 4 + 2], cy2 = sb[col * 4 + 3];
    int   clab = sl[col];
    float cArea = (cx2 - cx1) * (cy2 - cy1);

    // B-matrix 4x16 f32: VGPR0 = K0 row (lanes 0-15) / K2 (16-31);
    //                    VGPR1 = K1 / K3. Rows K2,K3 zero.
    v2f B;
    B.x = lo ? 1.0f  : 0.0f;
    B.y = lo ? cArea : 0.0f;
    v8f C = {};
    v8f D = __builtin_amdgcn_wmma_f32_16x16x4_f32(
        false, A, false, B, (short)0, C, false, false);

#pragma unroll
    for (int r = 0; r < 8; ++r) {
      int m = r + (lo ? 0 : 8);               // C/D layout: reg r -> M=r / M=r+8
      float iw = fminf(rbx[m][2], cx2) - fmaxf(rbx[m][0], cx1);
      float ih = fminf(rbx[m][3], cy2) - fmaxf(rbx[m][1], cy1);
      iw = fmaxf(iw, 0.0f); ih = fmaxf(ih, 0.0f);
      float inter = iw * ih;
      float asum = D[r];                      // area_m + area_n from WMMA
      int gi = rowBase + m;
      // iou > 0.5  <=>  inter > 0.5*(asum - inter + 1e-9)   (denominator > 0)
      bool pred = (clab == rlb[m]) && (col > gi) &&
                  (inter > 0.5f * (asum - inter + 1e-9f));
      u32 bal = __builtin_amdgcn_ballot_w32(pred);
      if (lane == r)     word |= (u64)(bal & 0xFFFFu) << (st * 16);
      if (lane == r + 8) word |= (u64)(bal >> 16)     << (st * 16);
    }
  }
  if (lane < 16)
    mask[(u64)(rowBase + lane) * NWORDS + cw] = word;
}

// ---------------------------------------------------------------------------
// 5) Sequential NMS reduce: single block, 128 threads each own one u64 of
//    the 8192-bit "removed" mask in LDS.
// ---------------------------------------------------------------------------
__global__ void __launch_bounds__(NWORDS) reduce_kernel(
    const u64* __restrict__ mask, const int* __restrict__ sv,
    int* __restrict__ keepArr) {
  __shared__ u64 removed[NWORDS];
  __shared__ int keepFlag;
  int tid = threadIdx.x;
  removed[tid] = 0ull;
  __syncthreads();
  for (int i = 0; i < M_ALL; ++i) {
    int wi = i >> 6, b = i & 63;
    if (tid == wi) {
      int k = (sv[i] != 0) && !((removed[wi] >> b) & 1ull);
      keepFlag = k;
      keepArr[i] = k;
    }
    __syncthreads();
    if (keepFlag) removed[tid] |= mask[(u64)i * NWORDS + tid];
    __syncthreads();
  }
}

// ---------------------------------------------------------------------------
// 6) Format outputs: (boxes[8192x4], labels[8192], scores[8192], keeps[8192])
//    flattened as floats in return order.
// ---------------------------------------------------------------------------
__global__ void final_kernel(const float* __restrict__ sb,
                             const int* __restrict__ sl,
                             const float* __restrict__ ss,
                             const int* __restrict__ keepArr,
                             float* __restrict__ out) {
  int i = blockIdx.x * blockDim.x + threadIdx.x;
  if (i >= M_ALL) return;
  int k = keepArr[i];
  float* bo = out;                    // 32768 floats
  float* lo = out + M_ALL * 4;        // +8192
  float* so = out + M_ALL * 5;        // +8192
  float* ko = out + M_ALL * 6;        // +8192
  bo[i * 4 + 0] = k ? sb[i * 4 + 0] : 0.0f;
  bo[i * 4 + 1] = k ? sb[i * 4 + 1] : 0.0f;
  bo[i * 4 + 2] = k ? sb[i * 4 + 2] : 0.0f;
  bo[i * 4 + 3] = k ? sb[i * 4 + 3] : 0.0f;
  lo[i] = k ? (float)sl[i] : -1.0f;
  so[i] = k ? ss[i] : 0.0f;
  ko[i] = k ? 1.0f : 0.0f;
}

// ---------------------------------------------------------------------------
extern "C" void kernel_launch(void* const* d_in, const int* in_sizes, int n_in,
                              void* d_out, int out_size, void* d_ws, size_t ws_size,
                              hipStream_t stream) {
  const float* boxes1  = (const float*)d_in[0];
  const float* boxes2  = (const float*)d_in[1];
  const float* scores1 = (const float*)d_in[2];
  const float* scores2 = (const float*)d_in[3];
  const int*   labels1 = (const int*)d_in[4];
  const int*   labels2 = (const int*)d_in[5];
  const int*   widthp  = (const int*)d_in[6];

  char* ws = (char*)d_ws;
  size_t off = 0;
  u64*   mask        = (u64*)(ws + off);   off += (size_t)M_ALL * NWORDS * 8;  // 8 MB
  float* boxesAll    = (float*)(ws + off); off += (size_t)M_ALL * 4 * 4;
  float* sortedBoxes = (float*)(ws + off); off += (size_t)M_ALL * 4 * 4;
  float* scoresAll   = (float*)(ws + off); off += (size_t)M_ALL * 4;
  float* sortedScores= (float*)(ws + off); off += (size_t)M_ALL * 4;
  int*   labelsAll   = (int*)(ws + off);   off += (size_t)M_ALL * 4;
  int*   sortedLabels= (int*)(ws + off);   off += (size_t)M_ALL * 4;
  int*   validAll    = (int*)(ws + off);   off += (size_t)M_ALL * 4;
  int*   sortedValid = (int*)(ws + off);   off += (size_t)M_ALL * 4;
  float* sArr        = (float*)(ws + off); off += (size_t)M_ALL * 4;
  u32*   order       = (u32*)(ws + off);   off += (size_t)M_ALL * 4;
  int*   keepArr     = (int*)(ws + off);   off += (size_t)M_ALL * 4;

  (void)in_sizes; (void)n_in; (void)out_size; (void)ws_size;

  prep_kernel<<<(M_ALL + 255) / 256, 256, 0, stream>>>(
      boxes1, boxes2, scores1, scores2, labels1, labels2, widthp,
      boxesAll, scoresAll, labelsAll, validAll, sArr);

  sort_kernel<<<1, SORT_THREADS, M_ALL * sizeof(u64), stream>>>(sArr, order);

  gather_kernel<<<(M_ALL + 255) / 256, 256, 0, stream>>>(
      order, boxesAll, scoresAll, labelsAll, validAll,
      sortedBoxes, sortedScores, sortedLabels, sortedValid);

  dim3 mgrid(NWORDS, M_ALL / 16);  // 128 col-words x 512 row-tiles
  mask_kernel<<<mgrid, 32, 0, stream>>>(sortedBoxes, sortedLabels, mask);

  reduce_kernel<<<1, NWORDS, 0, stream>>>(mask, sortedValid, keepArr);

  final_kernel<<<(M_ALL + 255) / 256, 256, 0, stream>>>(
      sortedBoxes, sortedLabels, sortedScores, keepArr, (float*)d_out);
}